// EMLTree1D_54700703482346
// MI455X (gfx1250) — compile-verified
//
#include <hip/hip_runtime.h>
#include <math.h>

// ---------------------------------------------------------------------------
// EMLTree1D on MI455X (gfx1250, wave32)
//
// batch=32768, vars=2, leaves=1024, internal=1023.
// - leaf values (the reference's `candidates @ w.T` GEMM) computed per wave
//   with V_WMMA_F32_16X16X4_F32, 16-leaf x 32-batch tiles, redistributed
//   through a per-wave bank-conflict-free LDS stage.
// - gate prob table staged to LDS with CDNA5 async global->LDS copies.
// - per-lane depth-first tree reduction, complex stack (depth<=10) in LDS.
// ---------------------------------------------------------------------------

#define N_VARS     2
#define N_LEAVES   1024
#define N_INTERNAL 1023
#define WG         128
#define NWAVES     (WG / 32)

#define EPSF    1.1920928955078125e-07f
#define BYPASSF (1.0f - EPSF)
#define CLAMPF  1e30f

// d_ws layout (floats):
//   [0,     4096) : leaf probs padded to 4/leaf   (1024 * 4)  [w0,w1,w2,0]
//   [4096, 12280) : gate probs (1023*2*4), natural order
#define WS_LEAF 0
#define WS_GATE 4096

typedef __attribute__((ext_vector_type(2))) float v2f;
typedef __attribute__((ext_vector_type(8))) float v8f;

__device__ __forceinline__ float clampv(float v) {
    v = __builtin_isnan(v) ? 0.0f : v;
    return fminf(CLAMPF, fmaxf(-CLAMPF, v));
}

// ps = (p_term, p_v0, p_v1, p_child); child = (cr, ci)
__device__ __forceinline__ void blend4(float4 g, float cr, float ci,
                                       float x0, float x1,
                                       float& br, float& bi) {
    float pt = g.x, p0 = g.y, p1 = g.z, pc = g.w;
    bool  mc  = pc > EPSF;                 // guard: zero BOTH p and child
    float crr = mc ? pc * cr : 0.0f;
    float cii = mc ? pc * ci : 0.0f;
    br = fmaf(p1, x1, fmaf(p0, x0, pt + crr));   // p_term*1 + child + var
    bi = cii;                                    // TERM_I=0, x imag=0
    if (pt > BYPASSF) { br = 1.0f; bi = 0.0f; }  // reference override order
    if (p0 > BYPASSF) { br = x0;   bi = 0.0f; }
    if (p1 > BYPASSF) { br = x1;   bi = 0.0f; }
}

// ---------------------------------------------------------------------------
// Prep: softmax leaf_logits (1024x3) and gate_logits (1023x2x4).
// ---------------------------------------------------------------------------
__global__ __launch_bounds__(WG) void prep_softmax(
        const float* __restrict__ leaf_logits,
        const float* __restrict__ gate_logits,
        float* __restrict__ ws,
        float* __restrict__ out, int batch) {
    int r = blockIdx.x * WG + threadIdx.x;
    float* out_leaf = out + 2 * batch;             // after complex level out
    float* out_gate = out_leaf + N_LEAVES * 3;
    if (r < N_LEAVES) {
        float a = leaf_logits[r * 3 + 0];
        float b = leaf_logits[r * 3 + 1];
        float c = leaf_logits[r * 3 + 2];
        float m = fmaxf(a, fmaxf(b, c));
        float ea = __expf(a - m), eb = __expf(b - m), ec = __expf(c - m);
        float inv = 1.0f / (ea + eb + ec);
        float p0 = ea * inv, p1 = eb * inv, p2 = ec * inv;
        ws[WS_LEAF + r * 4 + 0] = p0;
        ws[WS_LEAF + r * 4 + 1] = p1;
        ws[WS_LEAF + r * 4 + 2] = p2;
        ws[WS_LEAF + r * 4 + 3] = 0.0f;
        out_leaf[r * 3 + 0] = p0;
        out_leaf[r * 3 + 1] = p1;
        out_leaf[r * 3 + 2] = p2;
    } else if (r < N_LEAVES + 2 * N_INTERNAL) {
        int row = r - N_LEAVES;                    // 0..2045 = (node, side)
        const float* g = gate_logits + row * 4;
        float a = g[0], b = g[1], c = g[2], d = g[3];
        float m = fmaxf(fmaxf(a, b), fmaxf(c, d));
        float ea = __expf(a - m), eb = __expf(b - m);
        float ec = __expf(c - m), ed = __expf(d - m);
        float inv = 1.0f / (ea + eb + ec + ed);
        ws[WS_GATE + row * 4 + 0] = ea * inv;
        ws[WS_GATE + row * 4 + 1] = eb * inv;
        ws[WS_GATE + row * 4 + 2] = ec * inv;
        ws[WS_GATE + row * 4 + 3] = ed * inv;
        out_gate[row * 4 + 0] = ea * inv;
        out_gate[row * 4 + 1] = eb * inv;
        out_gate[row * 4 + 2] = ec * inv;
        out_gate[row * 4 + 3] = ed * inv;
    }
}

// ---------------------------------------------------------------------------
// Main kernel.
// ---------------------------------------------------------------------------
#define LVSTRIDE 17                       // pad 16 -> 17: conflict-free banks
#define LVWAVE   (32 * LVSTRIDE)          // floats per wave

__global__ __launch_bounds__(WG) void tree_eval(
        const float* __restrict__ x,
        const float* __restrict__ ws,
        float* __restrict__ out, int batch) {
    __shared__ float4 s_gate[2 * N_INTERNAL + 2];   // 32 KB
    __shared__ float  s_sr[10 * WG];                // 5 KB (stack real)
    __shared__ float  s_si[10 * WG];                // 5 KB (stack imag)
    __shared__ float  s_lv[NWAVES * LVWAVE];        // 8.5 KB leaf-tile stage

    const int tid  = threadIdx.x;
    const int lane = tid & 31;
    const int wave = tid >> 5;

    // ---- stage gate table into LDS via CDNA5 async global->LDS copies ----
    const float4* g_gate = (const float4*)(ws + WS_GATE);
    for (int i = tid; i < 2 * N_INTERNAL; i += WG) {
        unsigned lo = (unsigned)(size_t)(const void*)&s_gate[i];
        unsigned long long ga = (unsigned long long)(size_t)(g_gate + i);
        asm volatile("global_load_async_to_lds_b128 %0, %1, off"
                     :: "v"(lo), "v"(ga) : "memory");
    }
    asm volatile("s_wait_asynccnt 0" ::: "memory");
    __syncthreads();

    const int b = blockIdx.x * WG + tid;
    if (b >= batch) return;

    const float x0 = x[2 * b + 0];
    const float x1 = x[2 * b + 1];

    // ---- loop-invariant WMMA B operands (K=4 x N=16 batch columns) -------
    // B layout (f32 4x16, 2 VGPRs): VGPR0 = K0 (lanes 0-15) / K2 (16-31),
    //                               VGPR1 = K1 (lanes 0-15) / K3 (16-31).
    // column n = [1, x0[n], x1[n], 0]^T.
    const bool  lolane = (lane < 16);
    const float x0sw = __shfl_xor(x0, 16);
    const float x1sw = __shfl_xor(x1, 16);
    v2f B0, B1;                       // batches 0-15 of this wave
    B0[0] = lolane ? 1.0f : x1sw;     // K0 col=lane | K2 col=lane-16
    B0[1] = lolane ? x0   : 0.0f;     // K1 col=lane | K3 col=lane-16
    v2f B2, B3;                       // batches 16-31 of this wave
    B2[0] = lolane ? 1.0f : x1;       // K2 col n -> batch 16+n -> own x1
    B2[1] = lolane ? x0sw : 0.0f;     // K1 col n -> batch 16+n -> x0 of l+16
    (void)B1; (void)B3;

    const float4* g_leaf = (const float4*)(ws + WS_LEAF);
    const int wbase  = wave * LVWAVE;
    const int col    = lane & 15;
    const int rowoff = lolane ? 0 : 8;

    float vr = 0.0f, vi = 0.0f;
    int sp = 0;

    #pragma unroll 1
    for (int c = 0; c < N_LEAVES / 16; ++c) {
        // ---- A operand: 16 leaf rows [w0,w1,w2,0], K=4 (f32 16x4) -------
        // lanes 0-15: (K0,K1)=(w0,w1); lanes 16-31: (K2,K3)=(w2,0).
        float4 wv = g_leaf[c * 16 + col];
        v2f A;
        A[0] = lolane ? wv.x : wv.z;
        A[1] = lolane ? wv.y : 0.0f;

        v8f cz = {0.f, 0.f, 0.f, 0.f, 0.f, 0.f, 0.f, 0.f};
        // D[m][n] = w0[m] + w1[m]*x0[n] + w2[m]*x1[n]
        v8f d0 = __builtin_amdgcn_wmma_f32_16x16x4_f32(
                     false, A, false, B0, (short)0, cz, false, false);
        v8f d1 = __builtin_amdgcn_wmma_f32_16x16x4_f32(
                     false, A, false, B2, (short)0, cz, false, false);

        // ---- redistribute D tiles: (VGPR=leaf row, lane=batch col) ->
        //      per-lane (lane=batch) order via padded LDS stage ------------
        #pragma unroll
        for (int r = 0; r < 8; ++r) {
            s_lv[wbase + col * LVSTRIDE + rowoff + r]        = d0[r];
            s_lv[wbase + (16 + col) * LVSTRIDE + rowoff + r] = d1[r];
        }

        // ---- DFS over the 16 leaves of this chunk -----------------------
        #pragma unroll 1
        for (int k = 0; k < 16; ++k) {
            int leaf = c * 16 + k;
            vr = s_lv[wbase + lane * LVSTRIDE + k];   // leaf value (real)
            vi = 0.0f;

            int idx   = leaf;
            int base  = 0;      // gate base of current combine level
            int width = 512;    // #pairs at current combine level
            #pragma unroll 1
            while (idx & 1) {
                // pop left sibling; (vr,vi) is the right child
                --sp;
                float lr = s_sr[sp * WG + tid];
                float li = s_si[sp * WG + tid];

                int gi = base + (idx >> 1);            // uniform across wave
                float4 gl = s_gate[gi * 2 + 0];        // LDS broadcast
                float4 gr = s_gate[gi * 2 + 1];

                float blr, bli, brr, bri;
                blend4(gl, lr, li, x0, x1, blr, bli);  // left_in
                blend4(gr, vr, vi, x0, x1, brr, bri);  // right_in

                // exp(left_in) - log(right_in), complex
                float er = __expf(blr);
                float sn = __sinf(bli);
                float cs = __cosf(bli);
                float lm = __logf(hypotf(brr, bri));   // overflow-safe log|z|
                float ag = atan2f(bri, brr);

                vr = clampv(fmaf(er, cs, -lm));
                vi = clampv(fmaf(er, sn, -ag));

                base += width;
                width >>= 1;
                idx   >>= 1;
            }
            s_sr[sp * WG + tid] = vr;
            s_si[sp * WG + tid] = vi;
            ++sp;
        }
    }

    // after leaf 1023 the chain collapses to the root; it is in (vr, vi)
    out[2 * b + 0] = vr;
    out[2 * b + 1] = vi;
}

// ---------------------------------------------------------------------------
extern "C" void kernel_launch(void* const* d_in, const int* in_sizes, int n_in,
                              void* d_out, int out_size, void* d_ws, size_t ws_size,
                              hipStream_t stream) {
    const float* x  = (const float*)d_in[0];   // (batch, 2) f32
    const float* ll = (const float*)d_in[1];   // (1024, 3) f32
    const float* gl = (const float*)d_in[2];   // (1023, 2, 4) f32
    float* out = (float*)d_out;
    float* ws  = (float*)d_ws;

    const int batch = in_sizes[0] / N_VARS;

    const int prep_rows = N_LEAVES + 2 * N_INTERNAL;   // 3070
    prep_softmax<<<(prep_rows + WG - 1) / WG, WG, 0, stream>>>(ll, gl, ws, out, batch);
    tree_eval<<<(batch + WG - 1) / WG, WG, 0, stream>>>(x, ws, out, batch);
}